// PolylineSubgraphEncoder_670014898408
// MI455X (gfx1250) — compile-verified
//
#include <hip/hip_runtime.h>

// GCN 2-layer: N=100000 nodes, E=1000000 edges, HIDDEN=64, IN_DIM=4, fp32.
// GEMMs use V_WMMA_F32_16X16X4_F32 (fp32 exact). Edge pass uses
// global_atomic_add_f32 scatter (L2-resident: 25.6MB buffers << 192MB L2).

#define GCN_N 100000
#define GCN_E 1000000
#define GCN_H 64

typedef __attribute__((ext_vector_type(2))) float v2f;
typedef __attribute__((ext_vector_type(4))) float v4f;
typedef __attribute__((ext_vector_type(8))) float v8f;

// ---------------- utility kernels ----------------

__global__ __launch_bounds__(256) void fill1_kernel(float* __restrict__ p, int n) {
    int i = blockIdx.x * 256 + threadIdx.x;
    if (i < n) p[i] = 1.0f;
}

__global__ __launch_bounds__(256) void zero4_kernel(float* __restrict__ p, int n4) {
    int i = blockIdx.x * 256 + threadIdx.x;
    if (i < n4) ((v4f*)p)[i] = (v4f){0.f, 0.f, 0.f, 0.f};
}

__global__ __launch_bounds__(256) void deg_accum_kernel(const long long* __restrict__ dst,
                                                        float* __restrict__ deg, int nE) {
    int e = blockIdx.x * 256 + threadIdx.x;
    if (e < nE) atomicAdd(deg + (int)dst[e], 1.0f);
}

__global__ __launch_bounds__(256) void rsqrt_kernel(float* __restrict__ p, int n) {
    int i = blockIdx.x * 256 + threadIdx.x;
    if (i < n) p[i] = rsqrtf(p[i]);   // deg >= 1 always (self-loop), so no zero guard
}

// ---------------- layer 1 GEMM: h1 = x[100000,4] @ W1[4,64] ----------------
// One 16-row tile per wave, 2 waves per block. K=4 -> single WMMA step,
// 4 N-tiles -> 4 x v_wmma_f32_16x16x4_f32 per wave.
__global__ __launch_bounds__(64) void gemm1_kernel(const float* __restrict__ x,
                                                   const float* __restrict__ W1,
                                                   float* __restrict__ h1) {
    __shared__ float wlds[4 * 64];
    int t = threadIdx.x;
    ((v4f*)wlds)[t] = ((const v4f*)W1)[t];          // 64 thr * 16B = 1KB
    __syncthreads();

    int wave = t >> 5;
    int lane = t & 31;
    int tile = blockIdx.x * 2 + wave;               // 6250 tiles total, exact
    int m     = lane & 15;                          // row within tile
    int khalf = lane >> 4;                          // 0: K={0,1}, 1: K={2,3}
    int n     = lane & 15;                          // col within N-tile

    // A 16x4 f32 layout: lane m, VGPR0=K(2*khalf), VGPR1=K(2*khalf+1) -> contig b64 load
    v2f a = *(const v2f*)(x + (tile * 16 + m) * 4 + 2 * khalf);

    #pragma unroll
    for (int nb = 0; nb < 4; ++nb) {
        v2f b;
        b.x = wlds[(2 * khalf + 0) * 64 + nb * 16 + n];
        b.y = wlds[(2 * khalf + 1) * 64 + nb * 16 + n];
        v8f c = {};
        c = __builtin_amdgcn_wmma_f32_16x16x4_f32(false, a, false, b,
                                                  (short)0, c, false, false);
        #pragma unroll
        for (int r = 0; r < 8; ++r)
            h1[(tile * 16 + r + 8 * khalf) * 64 + nb * 16 + n] = c[r];
    }
}

// ---------------- layer 2 GEMM: h2 = h[100000,64] @ W2[64,64] ----------------
// W2 staged in LDS (16KB). 16 K-blocks x 4 N-tiles = 64 WMMAs per wave.
__global__ __launch_bounds__(64) void gemm2_kernel(const float* __restrict__ h,
                                                   const float* __restrict__ W2,
                                                   float* __restrict__ out) {
    __shared__ float wlds[64 * 64];
    int t = threadIdx.x;
    #pragma unroll
    for (int i = 0; i < 16; ++i)                    // 64 thr * 16 * 16B = 16KB
        ((v4f*)wlds)[t + i * 64] = ((const v4f*)W2)[t + i * 64];
    __syncthreads();

    int wave = t >> 5;
    int lane = t & 31;
    int tile = blockIdx.x * 2 + wave;
    int rowbase = tile * 16;
    int m     = lane & 15;
    int khalf = lane >> 4;
    int n     = lane & 15;

    v8f acc0 = {}, acc1 = {}, acc2 = {}, acc3 = {};

    #pragma unroll
    for (int kb = 0; kb < 16; ++kb) {
        int k0 = kb * 4 + 2 * khalf;
        v2f a = *(const v2f*)(h + (rowbase + m) * 64 + k0);   // contig b64 load

        v2f b0, b1, b2, b3;
        b0.x = wlds[k0 * 64 +  0 + n];  b0.y = wlds[k0 * 64 + 64 +  0 + n];
        b1.x = wlds[k0 * 64 + 16 + n];  b1.y = wlds[k0 * 64 + 64 + 16 + n];
        b2.x = wlds[k0 * 64 + 32 + n];  b2.y = wlds[k0 * 64 + 64 + 32 + n];
        b3.x = wlds[k0 * 64 + 48 + n];  b3.y = wlds[k0 * 64 + 64 + 48 + n];

        acc0 = __builtin_amdgcn_wmma_f32_16x16x4_f32(false, a, false, b0, (short)0, acc0, false, false);
        acc1 = __builtin_amdgcn_wmma_f32_16x16x4_f32(false, a, false, b1, (short)0, acc1, false, false);
        acc2 = __builtin_amdgcn_wmma_f32_16x16x4_f32(false, a, false, b2, (short)0, acc2, false, false);
        acc3 = __builtin_amdgcn_wmma_f32_16x16x4_f32(false, a, false, b3, (short)0, acc3, false, false);
    }

    #pragma unroll
    for (int r = 0; r < 8; ++r) {
        int row = rowbase + r + 8 * khalf;
        out[row * 64 +  0 + n] = acc0[r];
        out[row * 64 + 16 + n] = acc1[r];
        out[row * 64 + 32 + n] = acc2[r];
        out[row * 64 + 48 + n] = acc3[r];
    }
}

// ---------------- edge scatter: agg[dst] += h[src] * dinv[src]*dinv[dst] ----------------
// 8 lanes per edge, 8 floats per lane (2x b128 gather + 8 fp32 atomics).
__global__ __launch_bounds__(256) void edge_kernel(const long long* __restrict__ src,
                                                   const long long* __restrict__ dst,
                                                   const float* __restrict__ dinv,
                                                   const float* __restrict__ h,
                                                   float* __restrict__ agg, int nE) {
    int t = blockIdx.x * 256 + threadIdx.x;
    int e = t >> 3;
    if (e >= nE) return;
    int part = (t & 7) * 8;
    int s = (int)src[e];
    int d = (int)dst[e];
    float norm = dinv[s] * dinv[d];
    const float* hp = h + s * 64 + part;
    float*       ap = agg + d * 64 + part;
    v4f v0 = *(const v4f*)(hp);
    v4f v1 = *(const v4f*)(hp + 4);
    #pragma unroll
    for (int i = 0; i < 4; ++i) atomicAdd(ap + i,     v0[i] * norm);
    #pragma unroll
    for (int i = 0; i < 4; ++i) atomicAdd(ap + 4 + i, v1[i] * norm);
}

// ---------------- finalize: out = relu(agg + h*dinv^2 + bias) ----------------
// One thread per 4 elements (vectorized b128 traffic). Safe in-place (h==out ok).
__global__ __launch_bounds__(256) void finalize_kernel(const float* __restrict__ agg,
                                                       const float* __restrict__ h,
                                                       const float* __restrict__ dinv,
                                                       const float* __restrict__ bias,
                                                       float* __restrict__ out, int total4) {
    int t = blockIdx.x * 256 + threadIdx.x;
    if (t >= total4) return;
    int node = t >> 4;
    int d0   = (t & 15) * 4;
    float di  = dinv[node];
    float di2 = di * di;
    v4f av = *(const v4f*)(agg + t * 4);
    v4f hv = *(const v4f*)(h + t * 4);
    v4f bv = *(const v4f*)(bias + d0);
    v4f r;
    #pragma unroll
    for (int i = 0; i < 4; ++i)
        r[i] = fmaxf(av[i] + hv[i] * di2 + bv[i], 0.0f);
    *(v4f*)(out + t * 4) = r;
}

// ---------------- host launch ----------------

extern "C" void kernel_launch(void* const* d_in, const int* in_sizes, int n_in,
                              void* d_out, int out_size, void* d_ws, size_t ws_size,
                              hipStream_t stream) {
    const float*     x    = (const float*)d_in[0];
    const long long* edge = (const long long*)d_in[1];   // int64 [2, E]
    const float*     W1   = (const float*)d_in[2];
    const float*     b1   = (const float*)d_in[3];
    const float*     W2   = (const float*)d_in[4];
    const float*     b2   = (const float*)d_in[5];
    float*           out  = (float*)d_out;

    const int N = GCN_N, E = GCN_E;
    const long long* srcIdx = edge;
    const long long* dstIdx = edge + E;

    // Workspace layout (floats): dinv[padded 100352] | h1[N*64] | agg[N*64] ~= 51.6 MB
    float* dinv = (float*)d_ws;
    float* h1   = dinv + 100352;
    float* agg  = h1 + N * GCN_H;

    const int nodeBlocks  = (N + 255) / 256;            // 391
    const int edgeBlocks  = (E + 255) / 256;            // 3907
    const int scatBlocks  = (E * 8) / 256;              // 31250 (exact)
    const int featBlocks  = (N * GCN_H / 4) / 256;      // 6250 (exact) -> 1.6M vec4 threads
    const int gemmBlocks  = N / 32;                     // 3125 blocks * 2 waves * 16 rows

    // degree + symmetric norm (shared by both layers)
    fill1_kernel    <<<nodeBlocks, 256, 0, stream>>>(dinv, N);
    deg_accum_kernel<<<edgeBlocks, 256, 0, stream>>>(dstIdx, dinv, E);
    rsqrt_kernel    <<<nodeBlocks, 256, 0, stream>>>(dinv, N);

    // ---- layer 1 ----
    zero4_kernel   <<<featBlocks, 256, 0, stream>>>(agg, N * GCN_H / 4);
    gemm1_kernel   <<<gemmBlocks,  64, 0, stream>>>(x, W1, h1);
    edge_kernel    <<<scatBlocks, 256, 0, stream>>>(srcIdx, dstIdx, dinv, h1, agg, E);
    finalize_kernel<<<featBlocks, 256, 0, stream>>>(agg, h1, dinv, b1, h1, N * GCN_H / 4);

    // ---- layer 2 ----
    zero4_kernel   <<<featBlocks, 256, 0, stream>>>(agg, N * GCN_H / 4);
    gemm2_kernel   <<<gemmBlocks,  64, 0, stream>>>(h1, W2, out);       // d_out holds h2
    edge_kernel    <<<scatBlocks, 256, 0, stream>>>(srcIdx, dstIdx, dinv, out, agg, E);
    finalize_kernel<<<featBlocks, 256, 0, stream>>>(agg, out, dinv, b2, out, N * GCN_H / 4);
}